// BaseTransform_74749610819809
// MI455X (gfx1250) — compile-verified
//
#include <hip/hip_runtime.h>
#include <hip/hip_bf16.h>

// ---------------- problem constants ----------------
#define IMG_H 256
#define IMG_W 704
#define FH    32
#define FW    88
#define NDEP  118           // arange(1.0, 60.0, 0.5)
#define NCAM  6
#define NCH   80
#define NXX   360
#define NXY   360
#define NZv   1
#define NSPAT (NXX * NXY * NZv)          // 129600
#define NPC   (NDEP * FH * FW)           // 332288 points per camera (divisible by 16)
#define NPRIME (NCAM * NPC)              // 1993728
#define NTILES (NPRIME / 16)             // 124608
#define CAMP_FLOATS 128                  // padded cam-param region (512 B)

typedef __attribute__((ext_vector_type(2))) float v2f;
typedef __attribute__((ext_vector_type(8))) float v8f;

// ---------------- kernel 0: per-camera geometry setup ----------------
__device__ static void invert4x4(const float m[16], float out[16]) {
    float inv[16];
    inv[0]  =  m[5]*m[10]*m[15] - m[5]*m[11]*m[14] - m[9]*m[6]*m[15] + m[9]*m[7]*m[14] + m[13]*m[6]*m[11] - m[13]*m[7]*m[10];
    inv[4]  = -m[4]*m[10]*m[15] + m[4]*m[11]*m[14] + m[8]*m[6]*m[15] - m[8]*m[7]*m[14] - m[12]*m[6]*m[11] + m[12]*m[7]*m[10];
    inv[8]  =  m[4]*m[9]*m[15]  - m[4]*m[11]*m[13] - m[8]*m[5]*m[15] + m[8]*m[7]*m[13] + m[12]*m[5]*m[11] - m[12]*m[7]*m[9];
    inv[12] = -m[4]*m[9]*m[14]  + m[4]*m[10]*m[13] + m[8]*m[5]*m[14] - m[8]*m[6]*m[13] - m[12]*m[5]*m[10] + m[12]*m[6]*m[9];
    inv[1]  = -m[1]*m[10]*m[15] + m[1]*m[11]*m[14] + m[9]*m[2]*m[15] - m[9]*m[3]*m[14] - m[13]*m[2]*m[11] + m[13]*m[3]*m[10];
    inv[5]  =  m[0]*m[10]*m[15] - m[0]*m[11]*m[14] - m[8]*m[2]*m[15] + m[8]*m[3]*m[14] + m[12]*m[2]*m[11] - m[12]*m[3]*m[10];
    inv[9]  = -m[0]*m[9]*m[15]  + m[0]*m[11]*m[13] + m[8]*m[1]*m[15] - m[8]*m[3]*m[13] - m[12]*m[1]*m[11] + m[12]*m[3]*m[9];
    inv[13] =  m[0]*m[9]*m[14]  - m[0]*m[10]*m[13] - m[8]*m[1]*m[14] + m[8]*m[2]*m[13] + m[12]*m[1]*m[10] - m[12]*m[2]*m[9];
    inv[2]  =  m[1]*m[6]*m[15]  - m[1]*m[7]*m[14]  - m[5]*m[2]*m[15] + m[5]*m[3]*m[14] + m[13]*m[2]*m[7]  - m[13]*m[3]*m[6];
    inv[6]  = -m[0]*m[6]*m[15]  + m[0]*m[7]*m[14]  + m[4]*m[2]*m[15] - m[4]*m[3]*m[14] - m[12]*m[2]*m[7]  + m[12]*m[3]*m[6];
    inv[10] =  m[0]*m[5]*m[15]  - m[0]*m[7]*m[13]  - m[4]*m[1]*m[15] + m[4]*m[3]*m[13] + m[12]*m[1]*m[7]  - m[12]*m[3]*m[5];
    inv[14] = -m[0]*m[5]*m[14]  + m[0]*m[6]*m[13]  + m[4]*m[1]*m[14] - m[4]*m[2]*m[13] - m[12]*m[1]*m[6]  + m[12]*m[2]*m[5];
    inv[3]  = -m[1]*m[6]*m[11]  + m[1]*m[7]*m[10]  + m[5]*m[2]*m[11] - m[5]*m[3]*m[10] - m[9]*m[2]*m[7]   + m[9]*m[3]*m[6];
    inv[7]  =  m[0]*m[6]*m[11]  - m[0]*m[7]*m[10]  - m[4]*m[2]*m[11] + m[4]*m[3]*m[10] + m[8]*m[2]*m[7]   - m[8]*m[3]*m[6];
    inv[11] = -m[0]*m[5]*m[11]  + m[0]*m[7]*m[9]   + m[4]*m[1]*m[11] - m[4]*m[3]*m[9]  - m[8]*m[1]*m[7]   + m[8]*m[3]*m[5];
    inv[15] =  m[0]*m[5]*m[10]  - m[0]*m[6]*m[9]   - m[4]*m[1]*m[10] + m[4]*m[2]*m[9]  + m[8]*m[1]*m[6]   - m[8]*m[2]*m[5];
    float det = m[0]*inv[0] + m[1]*inv[4] + m[2]*inv[8] + m[3]*inv[12];
    float rdet = 1.0f / det;
    for (int i = 0; i < 16; ++i) out[i] = inv[i] * rdet;
}

__global__ void setup_kernel(const float* __restrict__ l2c,
                             const float* __restrict__ intr,
                             float* __restrict__ camP) {
    int t = threadIdx.x;
    if (t >= NCAM) {
        // zero the pad region so speculative reads in the scatter kernel are benign
        for (int i = NCAM * 12 + (t - NCAM); i < CAMP_FLOATS; i += (blockDim.x - NCAM))
            camP[i] = 0.0f;
        return;
    }
    float inv[16];
    invert4x4(l2c + t * 16, inv);

    const float* k = intr + t * 16;
    float k00=k[0],  k01=k[1],  k02=k[2];
    float k10=k[4],  k11=k[5],  k12=k[6];
    float k20=k[8],  k21=k[9],  k22=k[10];
    float det = k00*(k11*k22 - k12*k21) - k01*(k10*k22 - k12*k20) + k02*(k10*k21 - k11*k20);
    float id = 1.0f / det;
    float i00 =  (k11*k22 - k12*k21) * id;
    float i01 = -(k01*k22 - k02*k21) * id;
    float i02 =  (k01*k12 - k02*k11) * id;
    float i10 = -(k10*k22 - k12*k20) * id;
    float i11 =  (k00*k22 - k02*k20) * id;
    float i12 = -(k00*k12 - k02*k10) * id;
    float i20 =  (k10*k21 - k11*k20) * id;
    float i21 = -(k00*k21 - k01*k20) * id;
    float i22 =  (k00*k11 - k01*k10) * id;

    // combine = cam2lidar[:3,:3] @ K^-1 ; store rows [c0 c1 c2 | trans]
    for (int r = 0; r < 3; ++r) {
        float r0 = inv[r*4+0], r1 = inv[r*4+1], r2 = inv[r*4+2];
        camP[t*12 + r*4 + 0] = r0*i00 + r1*i10 + r2*i20;
        camP[t*12 + r*4 + 1] = r0*i01 + r1*i11 + r2*i21;
        camP[t*12 + r*4 + 2] = r0*i02 + r1*i12 + r2*i22;
        camP[t*12 + r*4 + 3] = inv[r*4+3];
    }
}

// ---------------- kernel 1: zero a float4 region ----------------
__global__ void zero_kernel(float4* __restrict__ p, long n4) {
    long i = (long)blockIdx.x * blockDim.x + threadIdx.x;
    long stride = (long)gridDim.x * blockDim.x;
    float4 z; z.x = 0.f; z.y = 0.f; z.z = 0.f; z.w = 0.f;
    for (; i < n4; i += stride) p[i] = z;
}

// ---------------- kernel 2: WMMA geometry + cooperative scatter ----------------
__global__ __launch_bounds__(256) void scatter_kernel(const float* __restrict__ x,
                                                      const float* __restrict__ camP,
                                                      float* __restrict__ dst,
                                                      long vstride, long cstride) {
    const int lane = threadIdx.x & 31;
    const int wave = (int)((blockIdx.x * blockDim.x + threadIdx.x) >> 5);
    const int nWaves = (int)((gridDim.x * blockDim.x) >> 5);
    const int m  = lane & 15;
    const int hi = lane >> 4;

    for (int tile = wave; tile < NTILES; tile += nWaves) {
        const int p0  = tile * 16;
        const int cam = p0 / NPC;          // all 16 points share a camera (NPC % 16 == 0)

        // prefetch next tile's 5120B feature block (global_prefetch_b8)
        int nextTile = tile + nWaves;
        if (nextTile < NTILES) {
            const char* pf = (const char*)(x + (size_t)nextTile * 16 * NCH);
            __builtin_prefetch(pf + (size_t)lane * 160, 0, 1);
        }

        // ---- A matrix (16x4 f32): rows 0..2 = [combine | trans], rest zero ----
        // layout: lanes 0-15 -> {K=0,K=1}, lanes 16-31 -> {K=2,K=3}
        const float* cp = camP + cam * 12;
        float sel = (m < 3) ? 1.0f : 0.0f;
        float a0 = sel * (hi ? cp[m*4 + 2] : cp[m*4 + 0]);
        float a1 = sel * (hi ? cp[m*4 + 3] : cp[m*4 + 1]);

        // ---- B matrix (4x16 f32): homogeneous frustum point (xs*d, ys*d, d, 1) ----
        int pidx = (p0 + m) % NPC;
        int di   = pidx / (FH * FW);
        int rem  = pidx - di * (FH * FW);
        int h    = rem / FW;
        int w    = rem - h * FW;
        float ds = 1.0f + 0.5f * (float)di;
        float xs = (float)w * (703.0f / 87.0f);   // linspace(0, IMG_W-1, FW)
        float ys = (float)h * (255.0f / 31.0f);   // linspace(0, IMG_H-1, FH)
        float b0 = hi ? ds   : xs * ds;
        float b1 = hi ? 1.0f : ys * ds;

        v2f a; a[0] = a0; a[1] = a1;
        v2f b; b[0] = b0; b[1] = b1;
        v8f c = {};
        // D(16x16) = A(16x4) x B(4x16): rows 0..2 = lidar-frame xyz of 16 points
        v8f g = __builtin_amdgcn_wmma_f32_16x16x4_f32(
            false, a, false, b, (short)0, c, false, false);

        // lanes 0-15 hold point N=lane: gx=VGPR0, gy=VGPR1, gz=VGPR2
        float gx = g[0], gy = g[1], gz = g[2];
        int cx = (int)((gx + 54.0f) / 0.3f);   // trunc toward zero == astype(int32)
        int cy = (int)((gy + 54.0f) / 0.3f);
        int cz = (int)((gz + 10.0f) / 20.0f);
        int kept = (cx >= 0) & (cx < NXX) & (cy >= 0) & (cy < NXY) & (cz >= 0) & (cz < NZv);
        int flat = (cz * NXX + cx) * NXY + cy;

        // ---- wave-cooperative scatter: 80 contiguous channels per point ----
        const float* src = x + (size_t)p0 * NCH;
        for (int p = 0; p < 16; ++p) {
            int kp = __shfl(kept, p, 32);
            int fp = __shfl(flat, p, 32);
            if (!kp) continue;
            const float* s = src + (size_t)p * NCH;
            float v0 = s[lane];
            float v1 = s[lane + 32];
            float* dv = dst + (size_t)fp * (size_t)vstride;
            atomicAdd(dv + (size_t)lane * (size_t)cstride, v0);
            atomicAdd(dv + (size_t)(lane + 32) * (size_t)cstride, v1);
            if (lane < 16) {
                float v2 = s[lane + 64];
                atomicAdd(dv + (size_t)(lane + 64) * (size_t)cstride, v2);
            }
        }
    }
}

// ---------------- kernel 3: (spatial, C) -> (C, spatial) transpose ----------------
__global__ __launch_bounds__(256) void transpose_kernel(const float* __restrict__ g,
                                                        float* __restrict__ out) {
    __shared__ float tile[64 * 81];           // odd pitch: conflict-free columns
    const int s0  = blockIdx.x * 64;          // NSPAT = 64 * 2025 exactly
    const int tid = threadIdx.x;
    #pragma unroll
    for (int it = 0; it < 20; ++it) {         // 64*80 = 5120 = 20*256
        int idx = it * 256 + tid;
        int sl = idx / 80, ch = idx - sl * 80;
        tile[sl * 81 + ch] = g[(size_t)(s0 + sl) * NCH + ch];
    }
    __syncthreads();
    #pragma unroll
    for (int it = 0; it < 20; ++it) {
        int idx = it * 256 + tid;
        int ch = idx >> 6;                    // /64
        int sl = idx & 63;
        out[(size_t)ch * NSPAT + s0 + sl] = tile[sl * 81 + ch];
    }
}

// ---------------- launcher ----------------
extern "C" void kernel_launch(void* const* d_in, const int* in_sizes, int n_in,
                              void* d_out, int out_size, void* d_ws, size_t ws_size,
                              hipStream_t stream) {
    (void)in_sizes; (void)n_in; (void)out_size;
    const float* x    = (const float*)d_in[0];
    const float* l2c  = (const float*)d_in[1];
    const float* intr = (const float*)d_in[2];
    float* out = (float*)d_out;
    float* wsf = (float*)d_ws;

    float* camP = wsf;                       // 512 B region
    float* grid = wsf + CAMP_FLOATS;         // channels-last (spatial, C) scratch grid
    const size_t gridFloats = (size_t)NSPAT * NCH;   // 10,368,000
    const bool useWs = ws_size >= (CAMP_FLOATS + gridFloats) * sizeof(float);

    setup_kernel<<<1, 64, 0, stream>>>(l2c, intr, camP);

    const long n4 = (long)(gridFloats / 4);
    if (useWs) {
        zero_kernel<<<1024, 256, 0, stream>>>((float4*)grid, n4);
        scatter_kernel<<<2048, 256, 0, stream>>>(x, camP, grid, 80L, 1L);
        transpose_kernel<<<NSPAT / 64, 256, 0, stream>>>(grid, out);
    } else {
        zero_kernel<<<1024, 256, 0, stream>>>((float4*)out, n4);
        scatter_kernel<<<2048, 256, 0, stream>>>(x, camP, out, 1L, (long)NSPAT);
    }
}